// GammaLambdaLearner_15367392985596
// MI455X (gfx1250) — compile-verified
//
#include <hip/hip_runtime.h>
#include <cstdint>

#define S_LEN 2048
#define SEG   8          // timesteps per thread
#define TPB   256        // threads per block (8 waves of 32)
#define WPB   (TPB / 32)
#define EPSV  1e-8f

typedef float v4f __attribute__((ext_vector_type(4)));   // real vector type
                                                         // (NT builtin needs it)

// ---------------------------------------------------------------------------
// prep: lam[t] = max(tanh(raw_lambd[t]), eps) into ws[0..S-1];
//       ws[S]  = max(tanh(raw_gamma), eps)
// ---------------------------------------------------------------------------
__global__ void glr_prep(const float* __restrict__ raw_gamma,
                         const float* __restrict__ raw_lambd,
                         float* __restrict__ ws) {
    int i = blockIdx.x * blockDim.x + threadIdx.x;
    if (i < S_LEN)  ws[i] = fmaxf(tanhf(raw_lambd[i]), EPSV);
    if (i == S_LEN) ws[S_LEN] = fmaxf(tanhf(raw_gamma[0]), EPSV);
}

// ---------------------------------------------------------------------------
// Main kernel: one block per batch row. Backward lambda-return scan.
//   ret[t] = a[t]*ret[t+1] + b[t],  a = g*(1-d)*lam,  b = r + g*(1-d)*(1-lam)*v
// Composite pairs (A,B) scanned across the block (suffix scan, time-reversed).
// ---------------------------------------------------------------------------
__global__ __launch_bounds__(TPB) void glr_scan(
    const float* __restrict__ values,   // [B, S+1]
    const float* __restrict__ rewards,  // [B, S]
    const float* __restrict__ dones,    // [B, S]
    const float* __restrict__ lam_g,    // ws: [S] lam, then [1] gamma
    float* __restrict__ out)            // [B, S]
{
    // Padded LDS tiles: +1 word per 64 words -> stride-8-per-lane reads are
    // bank-conflict-free on the 64-bank LDS.
    __shared__ float s_r[S_LEN + S_LEN / 64];
    __shared__ float s_d[S_LEN + S_LEN / 64];
    __shared__ float sWA[WPB], sWB[WPB];

    const int tid = threadIdx.x;
    const int row = blockIdx.x;
    const long rowRD = (long)row * S_LEN;        // rewards/dones/out row base
    const long rowV  = (long)row * (S_LEN + 1);  // values row base

    // ---- async stage rewards & dones into LDS (B128 = 16B chunk per lane) --
    {
        const unsigned lds_r = (unsigned)(uintptr_t)&s_r[0];
        const unsigned lds_d = (unsigned)(uintptr_t)&s_d[0];
        #pragma unroll
        for (int rep = 0; rep < 2; ++rep) {
            const int j = rep * TPB + tid;                 // chunk id 0..511
            const unsigned loff = (unsigned)(j * 16 + (j >> 4) * 4); // padded
            const uint64_t ga_r = (uint64_t)(uintptr_t)(rewards + rowRD + j * 4);
            const uint64_t ga_d = (uint64_t)(uintptr_t)(dones   + rowRD + j * 4);
            asm volatile("global_load_async_to_lds_b128 %0, %1, off"
                         :: "v"(lds_r + loff), "v"(ga_r) : "memory");
            asm volatile("global_load_async_to_lds_b128 %0, %1, off"
                         :: "v"(lds_d + loff), "v"(ga_d) : "memory");
        }
    }

    // ---- loads that overlap the async copy ---------------------------------
    const int t0 = tid * SEG;
    const float g = lam_g[S_LEN];                 // uniform -> scalar load
    const v4f L0 = *(const v4f*)(lam_g + t0);     // reused by all blocks: RT
    const v4f L1 = *(const v4f*)(lam_g + t0 + 4);
    const float lm[SEG] = {L0.x, L0.y, L0.z, L0.w, L1.x, L1.y, L1.z, L1.w};

    // values: single-use stream (row stride S+1 breaks 16B align -> b32),
    // non-temporal so 67MB of dead lines don't churn L2.
    float v[SEG];
    #pragma unroll
    for (int i = 0; i < SEG; ++i)
        v[i] = __builtin_nontemporal_load(values + rowV + t0 + 1 + i);
    const float vboot = __builtin_nontemporal_load(values + rowV + S_LEN);

    asm volatile("s_wait_asynccnt 0x0" ::: "memory");
    __syncthreads();

    // ---- per-element coefficients from LDS (conflict-free padded reads) ----
    const int wbase = t0 + (t0 >> 6);             // padded word index
    float a_[SEG], b_[SEG];
    #pragma unroll
    for (int i = 0; i < SEG; ++i) {
        const float r  = s_r[wbase + i];
        const float d  = s_d[wbase + i];
        const float gd = g * (1.0f - d);
        const float a  = gd * lm[i];
        a_[i] = a;
        b_[i] = fmaf(gd - a, v[i], r);            // r + g(1-d)(1-lam)v
    }

    // ---- segment composite: ret[t0] = A*c + B, c = ret[t0+SEG] -------------
    float A = 1.0f, Bc = 0.0f;
    #pragma unroll
    for (int i = SEG - 1; i >= 0; --i) {
        Bc = fmaf(a_[i], Bc, b_[i]);
        A  = a_[i] * A;
    }

    // ---- wave32 inclusive suffix scan of composites ------------------------
    const int lane = tid & 31;
    float IA = A, IB = Bc;
    #pragma unroll
    for (int off = 1; off < 32; off <<= 1) {
        float JA = __shfl_down(IA, off, 32);
        float JB = __shfl_down(IB, off, 32);
        const bool ok = (lane + off) < 32;
        JA = ok ? JA : 1.0f;
        JB = ok ? JB : 0.0f;
        IB = fmaf(IA, JB, IB);                    // I = I ∘ J (J applied first)
        IA = IA * JA;
    }

    const int wid = tid >> 5;
    if (lane == 0) { sWA[wid] = IA; sWB[wid] = IB; }
    __syncthreads();

    // E = composition of later waves (wid+1 .. WPB-1), outermost first
    float EA = 1.0f, EB = 0.0f;
    for (int w = wid + 1; w < WPB; ++w) {
        EB = fmaf(EA, sWB[w], EB);
        EA = EA * sWA[w];
    }

    // exclusive within-wave suffix: X = I_{lane+1}, identity at lane 31
    float XA = __shfl_down(IA, 1, 32);
    float XB = __shfl_down(IB, 1, 32);
    if (lane == 31) { XA = 1.0f; XB = 0.0f; }

    const float GA = XA * EA;
    const float GB = fmaf(XA, EB, XB);
    float c = fmaf(GA, vboot, GB);                // carry into this segment

    // ---- emit outputs (NT stores: 64MB single-use, keep L2 clean) ----------
    float o[SEG];
    #pragma unroll
    for (int i = SEG - 1; i >= 0; --i) {
        c = fmaf(a_[i], c, b_[i]);
        o[i] = c;
    }
    v4f* op = (v4f*)(out + rowRD + t0);
    const v4f o0 = {o[0], o[1], o[2], o[3]};
    const v4f o1 = {o[4], o[5], o[6], o[7]};
    __builtin_nontemporal_store(o0, op + 0);
    __builtin_nontemporal_store(o1, op + 1);
}

// ---------------------------------------------------------------------------
extern "C" void kernel_launch(void* const* d_in, const int* in_sizes, int n_in,
                              void* d_out, int out_size, void* d_ws, size_t ws_size,
                              hipStream_t stream) {
    const float* values    = (const float*)d_in[0];  // [B, S+1, 1]
    const float* rewards   = (const float*)d_in[1];  // [B, S, 1]
    const float* dones     = (const float*)d_in[2];  // [B, S, 1]
    const float* raw_gamma = (const float*)d_in[3];  // [1]
    const float* raw_lambd = (const float*)d_in[4];  // [S]
    float* out = (float*)d_out;
    float* ws  = (float*)d_ws;                       // needs (S+1) floats

    const int B = in_sizes[1] / S_LEN;               // 8192

    glr_prep<<<(S_LEN + 1 + TPB - 1) / TPB, TPB, 0, stream>>>(raw_gamma, raw_lambd, ws);
    glr_scan<<<B, TPB, 0, stream>>>(values, rewards, dones, ws, out);
}